// SKGModule_55224689492340
// MI455X (gfx1250) — compile-verified
//
#include <hip/hip_runtime.h>
#include <hip/hip_bf16.h>

typedef __attribute__((ext_vector_type(2))) float v2f;
typedef __attribute__((ext_vector_type(8))) float v8f;

#define NEG_SLOPE 0.01f

// ---------------------------------------------------------------------------
// fp32 WMMA GEMM, register-blocked: each wave computes a 32x32 output tile
// as 2x2 grid of 16x16 WMMA tiles (V_WMMA_F32_16X16X4_F32, K stepped by 4).
// Fragment reuse: per K-step, 2 A-frags + 2 B-frags feed 4 WMMAs.
// C[m,n] = sum_k A[m,k]*B[k,n] (+ colBias[n]).
// Assumes M,N multiples of 32, K multiple of 4 (true for all shapes here:
// M in {512, 20000}, N in {512, 256}, K in {512, 256}).
// ---------------------------------------------------------------------------
__global__ void wmma_gemm_f32(const float* __restrict__ A, long lda, long strideAb,
                              const float* __restrict__ B, long ldb, long strideBb,
                              float* __restrict__ C, long ldc, long strideCb,
                              const float* __restrict__ colBias, long strideBiasB,
                              int K)
{
    const int lane = threadIdx.x & 31;
    const int tm   = blockIdx.x * 32;
    const int tn   = blockIdx.y * 32;
    const int bz   = blockIdx.z;

    A += (long)bz * strideAb;
    B += (long)bz * strideBb;
    C += (long)bz * strideCb;

    const int half = lane >> 4;   // 0: K=k,k+1   1: K=k+2,k+3
    const int l15  = lane & 15;

    // A fragment: lane -> row; two row-tiles (tm, tm+16)
    const float* Ap0 = A + (long)(tm + l15) * lda + 2 * half;
    const float* Ap1 = Ap0 + 16 * lda;
    // B fragment: lane -> col; two col-tiles (tn, tn+16)
    const float* Bp0 = B + (long)(2 * half) * ldb + tn + l15;
    const float* Bp1 = Bp0 + 16;

    v8f acc00 = {}, acc01 = {}, acc10 = {}, acc11 = {};

    for (int k = 0; k < K; k += 4) {
        v2f a0, a1, b0, b1;
        a0.x = Ap0[0];   a0.y = Ap0[1];
        a1.x = Ap1[0];   a1.y = Ap1[1];
        b0.x = Bp0[0];   b0.y = Bp0[ldb];
        b1.x = Bp1[0];   b1.y = Bp1[ldb];
        acc00 = __builtin_amdgcn_wmma_f32_16x16x4_f32(false, a0, false, b0, (short)0, acc00, false, false);
        acc01 = __builtin_amdgcn_wmma_f32_16x16x4_f32(false, a0, false, b1, (short)0, acc01, false, false);
        acc10 = __builtin_amdgcn_wmma_f32_16x16x4_f32(false, a1, false, b0, (short)0, acc10, false, false);
        acc11 = __builtin_amdgcn_wmma_f32_16x16x4_f32(false, a1, false, b1, (short)0, acc11, false, false);
        Ap0 += 4;
        Ap1 += 4;
        Bp0 += 4 * (long)ldb;
        Bp1 += 4 * (long)ldb;
    }

    float cb0 = 0.0f, cb1 = 0.0f;
    if (colBias) {
        const float* cbp = colBias + (long)bz * strideBiasB;
        cb0 = cbp[tn + l15];
        cb1 = cbp[tn + 16 + l15];
    }

    // C/D layout: VGPR v -> row = v + half*8 (within 16x16 tile), col = lane&15
    const int r0 = tm + half * 8;
    const int r1 = tm + 16 + half * 8;
    const int c0 = tn + l15;
    const int c1 = tn + 16 + l15;
#pragma unroll
    for (int v = 0; v < 8; ++v) {
        C[(long)(r0 + v) * ldc + c0] = acc00[v] + cb0;
        C[(long)(r0 + v) * ldc + c1] = acc01[v] + cb1;
        C[(long)(r1 + v) * ldc + c0] = acc10[v] + cb0;
        C[(long)(r1 + v) * ldc + c1] = acc11[v] + cb1;
    }
}

// ---------------------------------------------------------------------------
// nodes_sum[n,f2] = nodes[n,0,f2] + nodes[n,1,f2]
// ---------------------------------------------------------------------------
__global__ void nodes_sum_kernel(const float* __restrict__ nodes,
                                 float* __restrict__ ns, long total, int F2)
{
    long i = (long)blockIdx.x * blockDim.x + threadIdx.x;
    if (i >= total) return;
    long n = i / F2;
    int  k = (int)(i - n * F2);
    const float* p = nodes + n * 2 * (long)F2;
    ns[i] = p[k] + p[F2 + k];
}

// ---------------------------------------------------------------------------
// c[row] = 2 * dot(x[row,:], bp)   (one wave per row)
// ---------------------------------------------------------------------------
__global__ void row_bias_dot(const float* __restrict__ x, const float* __restrict__ bp,
                             float* __restrict__ c, int rows, int F1)
{
    int warp = (int)((blockIdx.x * (long)blockDim.x + threadIdx.x) >> 5);
    int lane = threadIdx.x & 31;
    if (warp >= rows) return;
    const float* xr = x + (long)warp * F1;
    float s = 0.0f;
    for (int k = lane; k < F1; k += 32) s += xr[k] * bp[k];
#pragma unroll
    for (int off = 16; off; off >>= 1) s += __shfl_xor(s, off, 32);
    if (lane == 0) c[warp] = 2.0f * s;
}

// ---------------------------------------------------------------------------
// yT[b][f2][s] = y[(b*S+s)][f2]
// ---------------------------------------------------------------------------
__global__ void transpose_y(const float* __restrict__ y, float* __restrict__ yT,
                            long total, int S, int F2)
{
    long i = (long)blockIdx.x * blockDim.x + threadIdx.x;
    if (i >= total) return;
    int  s  = (int)(i % S);
    long t  = i / S;
    int  f2 = (int)(t % F2);
    int  b  = (int)(t / F2);
    yT[i] = y[((long)b * S + s) * F2 + f2];
}

// ---------------------------------------------------------------------------
// degree / dinv
// ---------------------------------------------------------------------------
__global__ void zero_f32(float* __restrict__ p, long n)
{
    long i = (long)blockIdx.x * blockDim.x + threadIdx.x;
    if (i < n) p[i] = 0.0f;
}

__global__ void count_deg(const int* __restrict__ dst, float* __restrict__ deg, int E)
{
    int e = blockIdx.x * blockDim.x + threadIdx.x;
    if (e < E) atomicAdd(&deg[dst[e]], 1.0f);
}

__global__ void make_dinv(const float* __restrict__ deg, float* __restrict__ dinv, int N)
{
    int n = blockIdx.x * blockDim.x + threadIdx.x;
    if (n < N) dinv[n] = rsqrtf(deg[n] + 1.0f);   // +1 self loop
}

// ---------------------------------------------------------------------------
// agg[b][n][s] = gcn_b_l[s] + hw[b][n][s] * dinv[n]^2   (self-loop folded in)
// ---------------------------------------------------------------------------
__global__ void init_agg(const float* __restrict__ hw, const float* __restrict__ dinv,
                         const float* __restrict__ bias, float* __restrict__ agg,
                         long total, int N, int S)
{
    long i = (long)blockIdx.x * blockDim.x + threadIdx.x;
    if (i >= total) return;
    int  s = (int)(i % S);
    long t = i / S;
    int  n = (int)(t % N);
    float d = dinv[n];
    agg[i] = bias[s] + hw[i] * d * d;
}

// ---------------------------------------------------------------------------
// Edge scatter: block of 256 threads handles 4 edges x 64 float4-chunks.
//   agg[b][dst][s..s+3] += hw[b][src][s..s+3] * dinv[src]*dinv[dst]
// agg (41 MB for both batches) fits in the 192 MB L2 -> atomics stay on-chip.
// ---------------------------------------------------------------------------
__global__ void scatter_edges(const int* __restrict__ ei, const float* __restrict__ hw,
                              const float* __restrict__ dinv, float* __restrict__ agg,
                              int E, int N, int S, int Bn)
{
    const int sc = threadIdx.x & 63;                   // S/4 = 64 chunks
    const int e  = blockIdx.x * 4 + (threadIdx.x >> 6);
    if (e >= E) return;
    int src = ei[e];
    int dst = ei[E + e];
    float nrm = dinv[src] * dinv[dst];
    for (int b = 0; b < Bn; ++b) {
        const float4 v = *(const float4*)(hw + ((long)b * N + src) * S + sc * 4);
        float* base = agg + ((long)b * N + dst) * S + sc * 4;
        atomicAdd(base + 0, v.x * nrm);
        atomicAdd(base + 1, v.y * nrm);
        atomicAdd(base + 2, v.z * nrm);
        atomicAdd(base + 3, v.w * nrm);
    }
}

// ---------------------------------------------------------------------------
// h = leaky_relu(agg)
// ---------------------------------------------------------------------------
__global__ void leaky_kernel(const float* __restrict__ agg, float* __restrict__ h, long total)
{
    long i = (long)blockIdx.x * blockDim.x + threadIdx.x;
    if (i >= total) return;
    float a = agg[i];
    h[i] = a > 0.0f ? a : NEG_SLOPE * a;
}

// ---------------------------------------------------------------------------
// out[b*N+n] = dot(h[b][n][:], w_bp) + b_bp[0]   (one wave per row)
// ---------------------------------------------------------------------------
__global__ void final_proj(const float* __restrict__ h, const float* __restrict__ w_bp,
                           const float* __restrict__ b_bp, float* __restrict__ out,
                           long rows, int S)
{
    long warp = ((long)blockIdx.x * blockDim.x + threadIdx.x) >> 5;
    int lane = threadIdx.x & 31;
    if (warp >= rows) return;
    const float* hr = h + warp * S;
    float s = 0.0f;
    for (int k = lane; k < S; k += 32) s += hr[k] * w_bp[k];
#pragma unroll
    for (int off = 16; off; off >>= 1) s += __shfl_xor(s, off, 32);
    if (lane == 0) out[warp] = s + b_bp[0];
}

// ---------------------------------------------------------------------------
extern "C" void kernel_launch(void* const* d_in, const int* in_sizes, int n_in,
                              void* d_out, int out_size, void* d_ws, size_t ws_size,
                              hipStream_t stream)
{
    const float* x     = (const float*)d_in[0];   // [B,S,F1]
    const float* nodes = (const float*)d_in[1];   // [N,M,F2]
    const int*   ei    = (const int*)d_in[2];     // [2,E]
    const float* Wp    = (const float*)d_in[3];   // [F1,F2]
    const float* bp    = (const float*)d_in[4];   // [F1]
    const float* gcn_W = (const float*)d_in[5];   // [L,S,S] (in,out)
    const float* gcn_b = (const float*)d_in[6];   // [L,S]
    const float* w_bp  = (const float*)d_in[7];   // [S]
    const float* b_bp  = (const float*)d_in[8];   // [1]
    float* out = (float*)d_out;                   // [B,N]

    const int Bn = 2, S = 256, F1 = 512, F2 = 512, M = 2, L = 3;
    const int E = in_sizes[2] / 2;                // 320000
    const int N = in_sizes[1] / (M * F2);         // 20000
    const int BS = Bn * S;                        // 512

    // ---- workspace carve-up (floats, 256B-aligned chunks) ----
    float* ws = (float*)d_ws;
    auto align = [](long v) { return (v + 63) & ~63L; };
    long off = 0;
    float* nodes_sum = ws + off; off += align((long)N * F2);      // 10.24M
    float* y         = ws + off; off += align((long)BS * F2);     // 0.26M
    float* yT        = ws + off; off += align((long)Bn * F2 * S); // 0.26M
    float* cvec      = ws + off; off += align(BS);
    float* deg       = ws + off; off += align(N);
    float* dinv      = ws + off; off += align(N);
    float* h         = ws + off; off += align((long)Bn * N * S);  // 10.24M
    float* hw        = ws + off; off += align((long)Bn * N * S);  // 10.24M
    float* agg       = ws + off; off += align((long)Bn * N * S);  // 10.24M
    (void)ws_size;

    const long hsz = (long)Bn * N * S;

    // 1) nodes_sum
    {
        long total = (long)N * F2;
        nodes_sum_kernel<<<(int)((total + 255) / 256), 256, 0, stream>>>(nodes, nodes_sum, total, F2);
    }

    // 2) y = x @ Wp   ([BS x F1] x [F1 x F2])
    {
        dim3 grid(BS / 32, F2 / 32, 1);
        wmma_gemm_f32<<<grid, 32, 0, stream>>>(x, F1, 0, Wp, F2, 0, y, F2, 0, nullptr, 0, F1);
    }

    // 3) cvec[bs] = 2 * dot(x[bs], bp)
    row_bias_dot<<<(BS * 32 + 255) / 256, 256, 0, stream>>>(x, bp, cvec, BS, F1);

    // 4) yT[b][f2][s] = y[b*S+s][f2]
    {
        long total = (long)Bn * F2 * S;
        transpose_y<<<(int)((total + 255) / 256), 256, 0, stream>>>(y, yT, total, S, F2);
    }

    // 5) h[b][n][s] = nodes_sum @ yT[b] + cvec[b*S+s]
    {
        dim3 grid(N / 32, S / 32, Bn);
        wmma_gemm_f32<<<grid, 32, 0, stream>>>(nodes_sum, F2, 0,
                                               yT, S, (long)F2 * S,
                                               h, S, (long)N * S,
                                               cvec, S, F2);
    }

    // 6) degrees -> dinv
    zero_f32<<<(N + 255) / 256, 256, 0, stream>>>(deg, N);
    count_deg<<<(E + 255) / 256, 256, 0, stream>>>(ei + E, deg, E);
    make_dinv<<<(N + 255) / 256, 256, 0, stream>>>(deg, dinv, N);

    // 7) GCN layers
    for (int l = 0; l < L; ++l) {
        // hw = h @ gcn_W[l]
        dim3 grid(N / 32, S / 32, Bn);
        wmma_gemm_f32<<<grid, 32, 0, stream>>>(h, S, (long)N * S,
                                               gcn_W + (long)l * S * S, S, 0,
                                               hw, S, (long)N * S,
                                               nullptr, 0, S);
        // agg = bias + self-loop contribution
        init_agg<<<(int)((hsz + 255) / 256), 256, 0, stream>>>(hw, dinv, gcn_b + (long)l * S,
                                                               agg, hsz, N, S);
        // edge scatter: 4 edges per 256-thread block
        scatter_edges<<<(E + 3) / 4, 256, 0, stream>>>(ei, hw, dinv, agg, E, N, S, Bn);
        // h = leaky_relu(agg)
        leaky_kernel<<<(int)((hsz + 255) / 256), 256, 0, stream>>>(agg, h, hsz);
    }

    // 8) out = h @ w_bp + b_bp
    {
        long rows = (long)Bn * N;
        final_proj<<<(int)((rows * 32 + 255) / 256), 256, 0, stream>>>(h, w_bp, b_bp, out, rows, S);
    }
}